// RBFNN_31413390803001
// MI455X (gfx1250) — compile-verified
//
#include <hip/hip_runtime.h>
#include <hip/hip_bf16.h>

// RBFNN L1-distance similarity for MI455X (gfx1250, wave32).
// Phase 1: tiled |x-c| L1 distance matrix -> d_out, global max -> d_ws (u32 bits).
// Phase 2: out = beta * (max*1.001 - sim), in place on d_out.
// CDNA5 paths: global_load_async_to_lds_b128 + s_wait_asynccnt double-buffered
// pipeline. Padded LDS rows (68 floats) give conflict-free ds_load_b128 with
// pure immediate offsets. 32x64 tiles -> 256 workgroups for WGP coverage.
// WMMA is mathematically inapplicable: L1 distance is not bilinear.

#define DIM    512             // inner dim == number of centers (both 512 here)
#define TX     32              // x-rows per block tile
#define TC     64              // centers per block tile
#define DK     64              // d-chunk per pipeline stage
#define KCH    (DIM / DK)      // 8 stages
#define ROWU   17              // 16B units per padded LDS row (64 floats + 4 pad)
#define XTILEU (TX * ROWU)     // 544 units: x tile
#define CTILEU (TC * ROWU)     // 1088 units: c tile
#define STAGEU (XTILEU + CTILEU) // 1632 units per stage
#define DWGT   1.001f          // distance_weight

__global__ void rbf_init_max(unsigned* gmax) { *gmax = 0u; }

__global__ __launch_bounds__(256) void rbf_l1_sim(
    const float* __restrict__ x, const float* __restrict__ c,
    float* __restrict__ sim, unsigned* __restrict__ gmax) {
  // ~51 KB: [2 stages][ x-tile | c-tile ], padded rows of 17 x 16B units
  __shared__ float4 smem[2 * STAGEU];

  const unsigned t  = threadIdx.x;
  const unsigned tx = t & 15u;   // center group (lane-varying within a wave)
  const unsigned ty = t >> 4;    // x-row group
  const unsigned xrow0 = blockIdx.x * TX;
  const unsigned crow0 = blockIdx.y * TC;

  // Low 32 bits of the flat shared address == LDS byte offset (ISA aperture rule).
  const unsigned lds0 = (unsigned)(size_t)(void*)&smem[0];

  // Per-thread async-copy assignment: 2 x-units + 4 c-units (16B each) per stage.
  unsigned ldsx[2], gx[2], ldsc[4], gc[4];
#pragma unroll
  for (int m = 0; m < 2; ++m) {
    const unsigned u   = t + 256u * m;          // 0..511 -> 32 rows x 16 qd
    const unsigned row = u >> 4, qd = u & 15u;
    ldsx[m] = lds0 + (row * ROWU + qd) * 16u;
    gx[m]   = (xrow0 + row) * (DIM * 4u) + qd * 16u;
  }
#pragma unroll
  for (int m = 0; m < 4; ++m) {
    const unsigned u   = t + 256u * m;          // 0..1023 -> 64 rows x 16 qd
    const unsigned row = u >> 4, qd = u & 15u;
    ldsc[m] = lds0 + (XTILEU + row * ROWU + qd) * 16u;
    gc[m]   = (crow0 + row) * (DIM * 4u) + qd * 16u;
  }

#define PREFETCH(KK, BUF)                                                      \
  {                                                                            \
    const unsigned go = (unsigned)(KK) * (DK * 4u);                            \
    const unsigned bo = (unsigned)(BUF) * (STAGEU * 16u);                      \
    _Pragma("unroll") for (int m = 0; m < 2; ++m) {                            \
      asm volatile("global_load_async_to_lds_b128 %0, %1, %2"                  \
                   :: "v"(ldsx[m] + bo), "v"(gx[m] + go), "s"(x) : "memory");  \
    }                                                                          \
    _Pragma("unroll") for (int m = 0; m < 4; ++m) {                            \
      asm volatile("global_load_async_to_lds_b128 %0, %1, %2"                  \
                   :: "v"(ldsc[m] + bo), "v"(gc[m] + go), "s"(c) : "memory");  \
    }                                                                          \
  }

#define COMPUTE(KK)                                                            \
  {                                                                            \
    const float4* __restrict__ xs = &smem[((KK) & 1) * STAGEU];                \
    const float4* __restrict__ cs = xs + XTILEU;                               \
    const float4* xp[2];                                                       \
    const float4* cp[4];                                                       \
    _Pragma("unroll") for (int j = 0; j < 2; ++j)                              \
      xp[j] = xs + (ty + 16u * j) * ROWU;                                      \
    _Pragma("unroll") for (int i = 0; i < 4; ++i)                              \
      cp[i] = cs + (tx + 16u * i) * ROWU;                                      \
    _Pragma("unroll") for (unsigned qd = 0; qd < 16; ++qd) {                   \
      float4 xv[2], cv[4];                                                     \
      _Pragma("unroll") for (int j = 0; j < 2; ++j) xv[j] = xp[j][qd];         \
      _Pragma("unroll") for (int i = 0; i < 4; ++i) cv[i] = cp[i][qd];         \
      _Pragma("unroll") for (int j = 0; j < 2; ++j)                            \
        _Pragma("unroll") for (int i = 0; i < 4; ++i) {                        \
          acc[j][i] += fabsf(xv[j].x - cv[i].x) + fabsf(xv[j].y - cv[i].y) +   \
                       fabsf(xv[j].z - cv[i].z) + fabsf(xv[j].w - cv[i].w);    \
        }                                                                      \
    }                                                                          \
  }

  float acc[2][4];
#pragma unroll
  for (int j = 0; j < 2; ++j)
#pragma unroll
    for (int i = 0; i < 4; ++i) acc[j][i] = 0.0f;

  PREFETCH(0, 0)

  // Steady state: prefetch k+1, wait for stage k (6 newer asyncs outstanding).
  for (int k = 0; k < KCH - 1; ++k) {
    PREFETCH(k + 1, (k + 1) & 1)
    asm volatile("s_wait_asynccnt 0x6" ::: "memory");
    __syncthreads();
    COMPUTE(k)
    __syncthreads();
  }
  // Last stage: drain all asyncs, no further prefetch or barrier needed after.
  asm volatile("s_wait_asynccnt 0x0" ::: "memory");
  __syncthreads();
  COMPUTE(KCH - 1)

#undef PREFETCH
#undef COMPUTE

  // Write sim tile (lanes 0..15 -> consecutive centers: coalesced) + local max.
  float mx = 0.0f;
#pragma unroll
  for (int j = 0; j < 2; ++j) {
    const unsigned r = xrow0 + ty + 16u * j;
#pragma unroll
    for (int i = 0; i < 4; ++i) {
      const unsigned o = crow0 + tx + 16u * i;
      sim[r * DIM + o] = acc[j][i];
      mx = fmaxf(mx, acc[j][i]);
    }
  }
  // wave32 max reduce, then one atomic per wave (sims >= 0: u32 order == f32 order)
  for (int off = 16; off > 0; off >>= 1) mx = fmaxf(mx, __shfl_xor(mx, off, 32));
  if ((t & 31u) == 0u) atomicMax(gmax, __float_as_uint(mx));
}

__global__ __launch_bounds__(256) void rbf_finalize(
    float* __restrict__ out, const float* __restrict__ beta,
    const unsigned* __restrict__ gmax, int n4) {
  const int i = blockIdx.x * blockDim.x + threadIdx.x;
  if (i >= n4) return;
  const float shift = __uint_as_float(*gmax) * DWGT;
  float4 v = ((const float4*)out)[i];
  const float4 b = ((const float4*)beta)[(unsigned)i & (DIM / 4u - 1u)];
  v.x = b.x * (shift - v.x);
  v.y = b.y * (shift - v.y);
  v.z = b.z * (shift - v.z);
  v.w = b.w * (shift - v.w);
  ((float4*)out)[i] = v;
}

extern "C" void kernel_launch(void* const* d_in, const int* in_sizes, int n_in,
                              void* d_out, int out_size, void* d_ws, size_t ws_size,
                              hipStream_t stream) {
  const float* x    = (const float*)d_in[0];   // [B, 512]
  const float* c    = (const float*)d_in[1];   // [512, 512]
  const float* beta = (const float*)d_in[2];   // [512]
  float*    out  = (float*)d_out;              // [B, 512]
  unsigned* gmax = (unsigned*)d_ws;            // 4 bytes of scratch

  const int B = in_sizes[0] / DIM;             // 1024
  const int O = in_sizes[2];                   // 512

  rbf_init_max<<<1, 1, 0, stream>>>(gmax);

  dim3 grid(B / TX, O / TC);                   // 32 x 8 = 256 workgroups
  rbf_l1_sim<<<grid, 256, 0, stream>>>(x, c, out, gmax);

  const int n4 = (B * O) / 4;
  rbf_finalize<<<(n4 + 255) / 256, 256, 0, stream>>>(out, beta, gmax, n4);
}